// HFNO2D_l2_overlap_4741643895639
// MI455X (gfx1250) — compile-verified
//
#include <hip/hip_runtime.h>
#include <math.h>

typedef __attribute__((ext_vector_type(2))) float v2f;
typedef __attribute__((ext_vector_type(8))) float v8f;

#define RES 64
#define DEPTH 32
#define BATCH 16
#define WIDTH 512
#define MAXNI 2112   // 64*33 upper bound
#define NEG_SLOPE 0.01f

// ---------------- workspace layout (in floats) ----------------
#define OFF_H   ((size_t)0)                              // 16*32*64*64      = 2097152
#define OFF_HR  (OFF_H  + (size_t)BATCH*DEPTH*RES*RES)   // 16*32*64*33*2    = 2162688
#define OFF_FX  (OFF_HR + (size_t)BATCH*DEPTH*RES*33*2)
#define OFF_T   (OFF_FX + (size_t)BATCH*DEPTH*RES*33*2)  // 16*64*2112       = 2162688
#define OFF_H1  (OFF_T  + (size_t)BATCH*64*MAXNI)        // 16*512
#define OFF_H2  (OFF_H1 + (size_t)BATCH*WIDTH)
#define OFF_OC  (OFF_H2 + (size_t)BATCH*WIDTH)           // 16*2*2112
#define OFF_S   (OFF_OC + (size_t)BATCH*2*MAXNI)         // 16*64*33*2
#define OFF_TT  (OFF_S  + (size_t)BATCH*RES*33*2)
#define OFF_IDX (OFF_TT + (size_t)BATCH*RES*33*2)        // ints: 2*3*2112

// ---------------- mask index build (matches np.nonzero row-major) -------
__global__ void k_build_indices(int* __restrict__ ii, int* __restrict__ jj) {
  if (threadIdx.x != 0 || blockIdx.x != 0) return;
  const int S2[3] = {0, 36, 196};
  const int E2[3] = {100, 324, 1024};
  for (int b = 0; b < 3; ++b) {
    int n = 0;
    for (int i = 0; i < 64; ++i) {
      int kx = (i < 32) ? i : i - 64;
      for (int j = 0; j < 33; ++j) {
        int ky = (j < 32) ? j : j - 64;
        int km2 = kx * kx + ky * ky;
        if (km2 >= S2[b] && km2 <= E2[b]) {
          ii[b * MAXNI + n] = i;
          jj[b * MAXNI + n] = j;
          ++n;
        }
      }
    }
  }
}

// ---------------- projection: h[b][c][r1][r2] = x@Pw + Pb ----------------
__global__ void k_proj(const float* __restrict__ x, const float* __restrict__ Pw,
                       const float* __restrict__ Pb, float* __restrict__ h) {
  int tid = blockIdx.x * blockDim.x + threadIdx.x;
  if (tid >= BATCH * RES * RES * DEPTH) return;
  int c  = tid & 31;
  int sp = tid >> 5;            // b*4096 + r1*64 + r2
  int r2 = sp & 63, r1 = (sp >> 6) & 63, b = sp >> 12;
  const float* xp = x + (size_t)sp * 3;
  float v = Pb[c] + xp[0] * Pw[0 * DEPTH + c] + xp[1] * Pw[1 * DEPTH + c]
                  + xp[2] * Pw[2 * DEPTH + c];
  h[(((size_t)(b * DEPTH + c) * RES + r1) * RES) + r2] = v;
}

// ---------------- rfft along r2: hr[row][j] (complex) --------------------
__global__ void k_rfft_r2(const float* __restrict__ h, float* __restrict__ hr) {
  int tid = blockIdx.x * blockDim.x + threadIdx.x;
  if (tid >= BATCH * DEPTH * RES * 33) return;
  int j = tid % 33, row = tid / 33;   // row = (b*32+c)*64 + r1
  const float* hp = h + (size_t)row * RES;
  float w = -2.0f * 3.14159265358979323846f * (float)j / (float)RES;
  float re = 0.f, im = 0.f;
  for (int r2 = 0; r2 < RES; ++r2) {
    float s, c;
    __sincosf(w * (float)r2, &s, &c);
    re += hp[r2] * c;
    im += hp[r2] * s;   // w negative -> im accumulates -sum(h*sin(theta))
  }
  hr[((size_t)row * 33 + j) * 2 + 0] = re;
  hr[((size_t)row * 33 + j) * 2 + 1] = im;
}

// ---------------- full complex FFT along r1: fx[b][c][i][j] --------------
__global__ void k_fft_r1(const float* __restrict__ hr, float* __restrict__ fx) {
  int tid = blockIdx.x * blockDim.x + threadIdx.x;
  if (tid >= BATCH * DEPTH * RES * 33) return;
  int j = tid % 33, i = (tid / 33) & 63, bc = tid / (33 * RES);
  const float* base = hr + ((size_t)bc * RES * 33 + j) * 2;
  float w = -2.0f * 3.14159265358979323846f * (float)i / (float)RES;
  float re = 0.f, im = 0.f;
  for (int r1 = 0; r1 < RES; ++r1) {
    float s, c;
    __sincosf(w * (float)r1, &s, &c);
    float xr = base[(size_t)r1 * 66 + 0];
    float xi = base[(size_t)r1 * 66 + 1];
    re += xr * c - xi * s;
    im += xr * s + xi * c;
  }
  fx[(((size_t)bc * RES + i) * 33 + j) * 2 + 0] = re;
  fx[(((size_t)bc * RES + i) * 33 + j) * 2 + 1] = im;
}

// ---------------- gather masked freqs into MLP input t -------------------
__global__ void k_gather(const float* __restrict__ fx, const int* __restrict__ ii,
                         const int* __restrict__ jj, float* __restrict__ t,
                         int band, int ni) {
  int tid = blockIdx.x * blockDim.x + threadIdx.x;
  if (tid >= BATCH * DEPTH * ni) return;
  int n = tid % ni, c = (tid / ni) % DEPTH, b = tid / (ni * DEPTH);
  int i = ii[band * MAXNI + n], j = jj[band * MAXNI + n];
  size_t src = (((size_t)(b * DEPTH + c) * RES + i) * 33 + j) * 2;
  size_t Kt = (size_t)64 * ni;
  t[(size_t)b * Kt + (size_t)c * 2 * ni + n]      = fx[src + 0];
  t[(size_t)b * Kt + (size_t)c * 2 * ni + ni + n] = fx[src + 1];
}

// ---------------- WMMA GEMM: out(16xN) = A(16xK) @ W(KxN) + bias ---------
// Block = 256 threads = 8 waves; each wave computes TWO 16-col tiles
// (32 cols), so one block covers 256 columns. The 16xKC A-chunk is staged
// in LDS once per block per chunk via the gfx1250 async-to-LDS path and
// shared by all 8 waves. K % 64 == 0 for all call sites.
// NSTATIC > 0: N known at compile time -> all W loads use immediate
// offsets (kk*N*4 fits the signed 24-bit field), only 4 pointer bumps
// per 64-K chunk. NSTATIC == 0: runtime N (ragged layer-2 GEMM).
#define KC 64
template <int NSTATIC>
__global__ void k_gemm16_wmma(const float* __restrict__ A, const float* __restrict__ W,
                              const float* __restrict__ bias, float* __restrict__ out,
                              int K, int Nrt, int relu) {
  const int N = (NSTATIC > 0) ? NSTATIC : Nrt;
  __shared__ float As[16][KC + 2];          // +2 pad: conflict-light ds_load_b64
  int lane = threadIdx.x & 31;
  int wv   = threadIdx.x >> 5;              // wave in block: 0..7
  int half = lane >> 4;                     // 0: lanes 0-15, 1: lanes 16-31
  int l16  = lane & 15;

  int colBase = (blockIdx.x * 8 + wv) * 32; // two 16-col tiles per wave
  int c0 = colBase + l16;
  int c1 = colBase + 16 + l16;
  int c0c = c0 < N ? c0 : N - 1;            // clamp: keeps EXEC all-ones
  int c1c = c1 < N ? c1 : N - 1;

  // lane-resident W pointers: rows (2*half) and (2*half+1), two col tiles
  const float* Q00 = W + (size_t)(half * 2) * N + c0c;
  const float* Q01 = Q00 + N;
  const float* Q10 = W + (size_t)(half * 2) * N + c1c;
  const float* Q11 = Q10 + N;
  const size_t chunkStride = (size_t)KC * N;

  // staging assignment: thread -> (row, 4-col group) of the A chunk
  int srow = threadIdx.x >> 4;              // 0..15
  int scol = (threadIdx.x & 15) * 4;        // 0,4,...,60
  unsigned ldsoff = (unsigned)(size_t)&As[srow][scol];

  v8f acc0 = {};
  v8f acc1 = {};
  for (int k = 0; k < K; k += KC) {
    __syncthreads();                        // previous chunk fully consumed
    {
      unsigned long long gaddr =
          (unsigned long long)(const void*)(A + (size_t)srow * K + k + scol);
      asm volatile("global_load_async_to_lds_b128 %0, %1, off"
                   :: "v"(ldsoff), "v"(gaddr) : "memory");
      asm volatile("s_wait_asynccnt 0x0" ::: "memory");
    }
    __syncthreads();                        // chunk visible to all waves

    if (k + 256 < K)                        // 4 chunks ahead
      __builtin_prefetch(Q00 + (size_t)256 * N, 0, 1);

#pragma unroll
    for (int kk = 0; kk < KC; kk += 4) {
      v2f a, b0, b1;
      a.x = As[l16][kk + half * 2 + 0];     // A: K = kb + 2*half (lane-half striping)
      a.y = As[l16][kk + half * 2 + 1];
      size_t o = (size_t)kk * N;            // immediate when NSTATIC > 0
      b0.x = Q00[o];
      b0.y = Q01[o];
      b1.x = Q10[o];
      b1.y = Q11[o];
      acc0 = __builtin_amdgcn_wmma_f32_16x16x4_f32(
          false, a, false, b0, (short)0, acc0, false, false);
      acc1 = __builtin_amdgcn_wmma_f32_16x16x4_f32(
          false, a, false, b1, (short)0, acc1, false, false);
    }
    Q00 += chunkStride;
    Q01 += chunkStride;
    Q10 += chunkStride;
    Q11 += chunkStride;
  }

  float bv0 = bias[c0c];
  float bv1 = bias[c1c];
#pragma unroll
  for (int r = 0; r < 8; ++r) {
    int row = half ? (r + 8) : r;
    float v0 = acc0[r] + bv0;
    float v1 = acc1[r] + bv1;
    if (relu) {
      v0 = v0 > 0.f ? v0 : NEG_SLOPE * v0;
      v1 = v1 > 0.f ? v1 : NEG_SLOPE * v1;
    }
    if (c0 < N) out[(size_t)row * N + c0] = v0;
    if (c1 < N) out[(size_t)row * N + c1] = v1;
  }
}

// ---------------- zero spectrum accumulator ------------------------------
__global__ void k_zero(float* __restrict__ p, int n) {
  int tid = blockIdx.x * blockDim.x + threadIdx.x;
  if (tid < n) p[tid] = 0.f;
}

// ---------------- scatter band coefficients into shared half-spectrum ----
__global__ void k_scatter(const float* __restrict__ oc, const int* __restrict__ ii,
                          const int* __restrict__ jj, float* __restrict__ S,
                          int band, int ni) {
  int tid = blockIdx.x * blockDim.x + threadIdx.x;
  if (tid >= BATCH * ni) return;
  int n = tid % ni, b = tid / ni;
  int i = ii[band * MAXNI + n], j = jj[band * MAXNI + n];
  float cr = oc[(size_t)b * 2 * ni + 2 * n + 0];
  float ci = oc[(size_t)b * 2 * ni + 2 * n + 1];
  size_t dst = (((size_t)b * RES + i) * 33 + j) * 2;
  S[dst + 0] += cr;    // bands run sequentially; within a band (b,i,j) unique
  S[dst + 1] += ci;
}

// ---------------- inverse FFT along kx: T[b][x][j] -----------------------
__global__ void k_ifft_kx(const float* __restrict__ S, float* __restrict__ T) {
  int tid = blockIdx.x * blockDim.x + threadIdx.x;
  if (tid >= BATCH * RES * 33) return;
  int j = tid % 33, x = (tid / 33) & 63, b = tid / (33 * RES);
  float w = 2.0f * 3.14159265358979323846f * (float)x / (float)RES;
  float re = 0.f, im = 0.f;
  for (int i = 0; i < RES; ++i) {
    float s, c;
    __sincosf(w * (float)i, &s, &c);
    float sr = S[(((size_t)b * RES + i) * 33 + j) * 2 + 0];
    float si = S[(((size_t)b * RES + i) * 33 + j) * 2 + 1];
    re += sr * c - si * s;
    im += sr * s + si * c;
  }
  T[(((size_t)b * RES + x) * 33 + j) * 2 + 0] = re;
  T[(((size_t)b * RES + x) * 33 + j) * 2 + 1] = im;
}

// ---------------- inverse real FFT along ky + final store ----------------
__global__ void k_ifft_ky(const float* __restrict__ T, float* __restrict__ out) {
  int tid = blockIdx.x * blockDim.x + threadIdx.x;
  if (tid >= BATCH * RES * RES) return;
  int y = tid & 63, x = (tid >> 6) & 63, b = tid >> 12;
  float w = 2.0f * 3.14159265358979323846f * (float)y / (float)RES;
  float acc = 0.f;
  for (int j = 0; j <= 32; ++j) {
    float s, c;
    __sincosf(w * (float)j, &s, &c);
    float tr = T[(((size_t)b * RES + x) * 33 + j) * 2 + 0];
    float ti = T[(((size_t)b * RES + x) * 33 + j) * 2 + 1];
    float beta = (j == 0 || j == 32) ? 1.f : 2.f;
    acc += beta * (tr * c - ti * s);
  }
  out[((size_t)(b * RES + x) * RES) + y] = acc * (1.0f / 4096.0f);
}

// ---------------- host side ----------------------------------------------
static inline int cdiv(long n, int b) { return (int)((n + b - 1) / b); }

extern "C" void kernel_launch(void* const* d_in, const int* in_sizes, int n_in,
                              void* d_out, int out_size, void* d_ws, size_t ws_size,
                              hipStream_t stream) {
  (void)in_sizes; (void)n_in; (void)out_size; (void)ws_size;
  const float* x  = (const float*)d_in[0];
  const float* Pw = (const float*)d_in[1];
  const float* Pb = (const float*)d_in[2];
  float* out = (float*)d_out;
  float* ws  = (float*)d_ws;

  // band sizes, identical to device mask (exact integer-squared compares)
  int ni[3];
  {
    const int S2[3] = {0, 36, 196}, E2[3] = {100, 324, 1024};
    for (int b = 0; b < 3; ++b) ni[b] = 0;
    for (int i = 0; i < 64; ++i) {
      int kx = (i < 32) ? i : i - 64;
      for (int j = 0; j < 33; ++j) {
        int ky = (j < 32) ? j : j - 64;
        int km2 = kx * kx + ky * ky;
        for (int b = 0; b < 3; ++b)
          if (km2 >= S2[b] && km2 <= E2[b]) ni[b]++;
      }
    }
  }

  float* h  = ws + OFF_H;
  float* hr = ws + OFF_HR;
  float* fx = ws + OFF_FX;
  float* t  = ws + OFF_T;
  float* h1 = ws + OFF_H1;
  float* h2 = ws + OFF_H2;
  float* oc = ws + OFF_OC;
  float* S  = ws + OFF_S;
  float* T  = ws + OFF_TT;
  int*   ii = (int*)(ws + OFF_IDX);
  int*   jj = ii + 3 * MAXNI;

  k_build_indices<<<1, 1, 0, stream>>>(ii, jj);
  k_zero<<<cdiv(BATCH * RES * 33 * 2, 256), 256, 0, stream>>>(S, BATCH * RES * 33 * 2);

  k_proj<<<cdiv((long)BATCH * RES * RES * DEPTH, 256), 256, 0, stream>>>(x, Pw, Pb, h);
  k_rfft_r2<<<cdiv((long)BATCH * DEPTH * RES * 33, 256), 256, 0, stream>>>(h, hr);
  k_fft_r1<<<cdiv((long)BATCH * DEPTH * RES * 33, 256), 256, 0, stream>>>(hr, fx);

  for (int band = 0; band < 3; ++band) {
    int n = ni[band];
    const float* W0 = (const float*)d_in[3 + band * 6 + 0];
    const float* b0 = (const float*)d_in[3 + band * 6 + 1];
    const float* W1 = (const float*)d_in[3 + band * 6 + 2];
    const float* b1 = (const float*)d_in[3 + band * 6 + 3];
    const float* W2 = (const float*)d_in[3 + band * 6 + 4];
    const float* b2 = (const float*)d_in[3 + band * 6 + 5];

    k_gather<<<cdiv((long)BATCH * DEPTH * n, 256), 256, 0, stream>>>(fx, ii, jj, t, band, n);

    // grid: each block covers 256 output columns (8 waves x 2 tiles x 16)
    int K0 = 64 * n;                        // multiple of 64
    int blk512 = cdiv(WIDTH, 256);          // = 2
    k_gemm16_wmma<WIDTH><<<blk512, 256, 0, stream>>>(t,  W0, b0, h1, K0,   WIDTH, 1);
    k_gemm16_wmma<WIDTH><<<blk512, 256, 0, stream>>>(h1, W1, b1, h2, WIDTH, WIDTH, 1);

    int N2 = 2 * n;
    k_gemm16_wmma<0><<<cdiv(N2, 256), 256, 0, stream>>>(h2, W2, b2, oc, WIDTH, N2, 0);

    k_scatter<<<cdiv((long)BATCH * n, 256), 256, 0, stream>>>(oc, ii, jj, S, band, n);
  }

  k_ifft_kx<<<cdiv((long)BATCH * RES * 33, 256), 256, 0, stream>>>(S, T);
  k_ifft_ky<<<cdiv((long)BATCH * RES * RES, 256), 256, 0, stream>>>(T, out);
}